// PermutohedralEncoding_18605798326297
// MI455X (gfx1250) — compile-verified
//
#include <hip/hip_runtime.h>
#include <stdint.h>

// Permutohedral hash-grid encoding for MI455X (gfx1250, wave32).
// L2-gather + VALU bound: the 48 MB of lattice tables are L2-resident
// (192 MB L2), and 48M random 8B gathers dominate. No shared-operand matmul
// structure -> WMMA inapplicable; the CDNA5-specific paths used are the
// async global->LDS staging (ASYNCcnt), global_prefetch_b8, LDS broadcast
// reads, and wave32-sized blocks.

namespace {
constexpr int      kLevels   = 24;
constexpr int      kFeat     = 2;
constexpr uint32_t kCapacity = 1u << 18;
constexpr uint32_t kCapMask  = kCapacity - 1u;
constexpr int      kBlock    = 256;
constexpr int      kLvlPad   = 8;   // floats per level in LDS (32 B, b128-friendly)
}

__global__ __launch_bounds__(kBlock)
void permuto_encode_kernel(const float* __restrict__ positions,   // [N,3]
                           const float* __restrict__ lattice,     // [24, 2^18, 2]
                           const float* __restrict__ shift,       // [24,3]
                           const float* __restrict__ scale,       // [24,3]
                           float* __restrict__ out,               // [N, 48]
                           int N)
{
    // Per-level constants, padded to 32 B/level:
    // [sh0, sh1, sh2, sc0, sc1, sc2, pad, pad]
    __shared__ __attribute__((aligned(16))) float s_const[kLevels * kLvlPad];

    const int tid = threadIdx.x;
    const int n   = blockIdx.x * kBlock + tid;

    // ---- Stage level constants into LDS via CDNA5 async global->LDS loads.
    // Lanes 0..71 each move one dword of shift and one of scale into the
    // padded layout.
    if (tid < kLevels * 3) {
        const int l = tid / 3;
        const int i = tid - 3 * l;
        const unsigned base = (unsigned)(unsigned long long)(&s_const[0]); // LDS byte offset
        const unsigned dst_sh = base + (unsigned)(l * kLvlPad + i) * 4u;
        const unsigned dst_sc = dst_sh + 12u;
        const unsigned voff   = (unsigned)(tid * 4);
        asm volatile("global_load_async_to_lds_b32 %0, %1, %2"
                     :: "v"(dst_sh), "v"(voff), "s"(shift) : "memory");
        asm volatile("global_load_async_to_lds_b32 %0, %1, %2"
                     :: "v"(dst_sc), "v"(voff), "s"(scale) : "memory");
    }

    // Overlap: pull this thread's position line into the near caches while
    // the async staging is in flight (WGP-scope global_prefetch_b8).
    if (n < N) __builtin_prefetch(positions + 3 * (size_t)n, 0, 3);

    // Issuing waves drain their ASYNCcnt before signaling the barrier;
    // non-issuing waves have ASYNCcnt==0 (no-op wait).
    asm volatile("s_wait_asynccnt 0" ::: "memory");
    __syncthreads();

    if (n >= N) return;

    const float x0 = positions[3 * (size_t)n + 0];
    const float x1 = positions[3 * (size_t)n + 1];
    const float x2 = positions[3 * (size_t)n + 2];

    float* outp = out + (size_t)n * (kLevels * kFeat);

    for (int l = 0; l < kLevels; ++l) {
        // One b128 + one b64 LDS broadcast per level.
        const float4 c0 = *(const float4*)(&s_const[l * kLvlPad]);      // sh0 sh1 sh2 sc0
        const float2 c1 = *(const float2*)(&s_const[l * kLvlPad + 4]);  // sc1 sc2

        const float cf0 = (x0 + c0.x) * c0.w;
        const float cf1 = (x1 + c0.y) * c1.x;
        const float cf2 = (x2 + c0.z) * c1.y;

        // elevate: reverse cumsum then subtract i*cf
        const float S2 = cf2;
        const float S1 = cf1 + cf2;
        const float S0 = cf0 + S1;
        float e[4];
        e[0] = S0;
        e[1] = S1 - cf0;
        e[2] = S2 - 2.0f * cf1;
        e[3] = -3.0f * cf2;

        // nearest multiple of 4 (ties -> down, matching strict '<' in ref)
        float rem0f[4];
        float sumr = 0.0f;
#pragma unroll
        for (int i = 0; i < 4; ++i) {
            const float v  = e[i] * 0.25f;
            const float dn = floorf(v) * 4.0f;
            const float up = ceilf(v) * 4.0f;
            const float r  = ((up - e[i]) < (e[i] - dn)) ? up : dn;
            rem0f[i] = r;
            sumr += r;
        }
        const int sum_i = (int)rintf(sumr * 0.25f);

        float diff[4];
#pragma unroll
        for (int i = 0; i < 4; ++i) diff[i] = e[i] - rem0f[i];

        int rank[4];
#pragma unroll
        for (int i = 0; i < 4; ++i) {
            int rk = 0;
#pragma unroll
            for (int j = 0; j < 4; ++j) {
                rk += ((diff[j] > diff[i]) || ((diff[j] == diff[i]) && (j < i))) ? 1 : 0;
            }
            rank[i] = rk + sum_i;
        }

        int rem0i[4];
#pragma unroll
        for (int i = 0; i < 4; ++i) {
            int ri = (int)rintf(rem0f[i]);
            int rk = rank[i];
            if (rk < 0) { rk += 4; ri += 4; }
            if (rk > 3) { rk -= 4; ri -= 4; }
            rank[i]  = rk;
            rem0i[i] = ri;
        }

        float delta[4];
#pragma unroll
        for (int i = 0; i < 4; ++i) delta[i] = (e[i] - (float)rem0i[i]) * 0.25f;

        // barycentric weights via unrolled selects (no dynamic local indexing)
        float b0 = 0.f, b1 = 0.f, b2 = 0.f, b3 = 0.f, b4 = 0.f;
#pragma unroll
        for (int i = 0; i < 4; ++i) {
            const int   ia = 3 - rank[i];
            const float d  = delta[i];
            b0 += (ia == 0) ? d : 0.f;
            b1 += (ia == 1) ? d : 0.f;  b1 -= (ia == 0) ? d : 0.f;
            b2 += (ia == 2) ? d : 0.f;  b2 -= (ia == 1) ? d : 0.f;
            b3 += (ia == 3) ? d : 0.f;  b3 -= (ia == 2) ? d : 0.f;
            b4 -= (ia == 3) ? d : 0.f;
        }
        b0 += 1.0f + b4;
        const float bw[4] = { b0, b1, b2, b3 };

        // 4 vertex gathers from the (L2-resident) level table
        const float* tbl = lattice + (((size_t)l) << 18) * kFeat;
        float a0 = 0.0f, a1 = 0.0f;
#pragma unroll
        for (int v = 0; v < 4; ++v) {
            const uint32_t k0 = (uint32_t)(rem0i[0] + v - ((rank[0] > 3 - v) ? 4 : 0));
            const uint32_t k1 = (uint32_t)(rem0i[1] + v - ((rank[1] > 3 - v) ? 4 : 0));
            const uint32_t k2 = (uint32_t)(rem0i[2] + v - ((rank[2] > 3 - v) ? 4 : 0));
            const uint32_t h    = k0 ^ (k1 * 2654435761u) ^ (k2 * 805459861u);
            const uint32_t slot = h & kCapMask;
            const float2 f = *(const float2*)(tbl + (size_t)slot * kFeat);
            a0 = fmaf(bw[v], f.x, a0);
            a1 = fmaf(bw[v], f.y, a1);
        }

        *(float2*)(outp + l * kFeat) = make_float2(a0, a1);
    }
}

extern "C" void kernel_launch(void* const* d_in, const int* in_sizes, int n_in,
                              void* d_out, int out_size, void* d_ws, size_t ws_size,
                              hipStream_t stream) {
    const float* positions = (const float*)d_in[0];
    const float* lattice   = (const float*)d_in[1];
    const float* shift     = (const float*)d_in[2];
    const float* scale     = (const float*)d_in[3];
    float*       out       = (float*)d_out;

    const int N    = in_sizes[0] / 3;
    const int grid = (N + kBlock - 1) / kBlock;

    hipLaunchKernelGGL(permuto_encode_kernel, dim3(grid), dim3(kBlock), 0, stream,
                       positions, lattice, shift, scale, out, N);
}